// QueryAndGroup_52896817217695
// MI455X (gfx1250) — compile-verified
//
#include <hip/hip_runtime.h>

#define B_ 8
#define N_ 8192
#define S_ 2048
#define C_ 64
#define NS_ 32
#define R2_ 0.04f
#define QT 16          // queries per block
#define CHUNK 2048     // points per LDS predicate chunk
#define NCHUNK (N_ / CHUNK)

typedef __attribute__((ext_vector_type(2))) float v2f;
typedef __attribute__((ext_vector_type(8))) float v8f;

__global__ __launch_bounds__(256)
void qg_ballquery_wmma(const float* __restrict__ xyz,
                       const float* __restrict__ new_xyz,
                       const float* __restrict__ feat,
                       float* __restrict__ out)
{
    __shared__ float qx[QT], qy[QT], qz[QT], qn[QT];
    // transposed predicate buffer: predT[point_local][q], 16 bytes per point (32 KB)
    __shared__ unsigned char predT[CHUNK * QT];
    __shared__ int selIdx[QT][NS_];
    __shared__ int selCnt[QT];
    __shared__ int nfull;

    const int tid  = threadIdx.x;
    const int wave = tid >> 5;
    const int lane = tid & 31;
    const int b    = blockIdx.x >> 7;            // S_/QT = 128 tiles per batch
    const int s0   = (blockIdx.x & 127) * QT;

    // ---- load the 16 query points for this tile
    if (tid < QT) {
        const float* qp = new_xyz + ((size_t)b * S_ + s0 + tid) * 3;
        float x = qp[0], y = qp[1], z = qp[2];
        qx[tid] = x; qy[tid] = y; qz[tid] = z;
        qn[tid] = x * x + y * y + z * z;
        selCnt[tid] = 0;
        if (tid == 0) nfull = 0;
    }
    __syncthreads();

    // ---- A operand: row q = (-2qx, -2qy, -2qz, |q|^2), K split across lane halves
    const int  qs = lane & 15;
    const bool hi = lane >= 16;
    v2f amat;
    amat.x = hi ? (-2.0f * qz[qs]) : (-2.0f * qx[qs]);
    amat.y = hi ? qn[qs]           : (-2.0f * qy[qs]);

    int cnt0 = 0, cnt1 = 0;                      // selection counters (wave-uniform)
    const unsigned lt = (1u << lane) - 1u;

    for (int chunk = 0; chunk < NCHUNK; ++chunk) {
        const int cbase = chunk * CHUNK;

        // ===== phase A: WMMA distance predicates =====
        for (int i = 0; i < 16; ++i) {
            const int pbase = cbase + wave * 256 + i * 16;
            const int pidx  = pbase + qs;
            const float* pp = xyz + ((size_t)b * N_ + pidx) * 3;
            float px = pp[0], py = pp[1], pz = pp[2];
            float thr = R2_ - (px * px + py * py + pz * pz);  // d^2<r^2  <=>  c < thr

            // B operand: col p = (px, py, pz, 1), K split across lane halves
            v2f bmat;
            bmat.x = hi ? pz   : px;
            bmat.y = hi ? 1.0f : py;

            v8f c = {};
            // c[q][p] = |q|^2 - 2 q.p   (16 queries x 16 points)
            c = __builtin_amdgcn_wmma_f32_16x16x4_f32(
                    false, amat, false, bmat, (short)0, c, false, false);

            // pack this lane's 8 predicate bytes (queries v / v+8 for its point)
            unsigned w0 = (c[0] < thr ? 0x00000001u : 0u)
                        | (c[1] < thr ? 0x00000100u : 0u)
                        | (c[2] < thr ? 0x00010000u : 0u)
                        | (c[3] < thr ? 0x01000000u : 0u);
            unsigned w1 = (c[4] < thr ? 0x00000001u : 0u)
                        | (c[5] < thr ? 0x00000100u : 0u)
                        | (c[6] < thr ? 0x00010000u : 0u)
                        | (c[7] < thr ? 0x01000000u : 0u);
            const int lpoint = wave * 256 + i * 16 + qs;     // chunk-local point
            uint2 w; w.x = w0; w.y = w1;
            // one conflict-free b64 store: bytes [q..q+7] of row lpoint
            *(uint2*)(predT + lpoint * QT + (hi ? 8 : 0)) = w;
        }
        __syncthreads();

        // ===== phase B: ordered first-32 selection via ballot compaction =====
        // wave w owns queries 2w and 2w+1
        for (int sub = 0; sub < 2; ++sub) {
            int q   = wave * 2 + sub;
            int cnt = sub ? cnt1 : cnt0;
            const int was = cnt;
            if (cnt < NS_) {
                for (int it = 0; it < CHUNK / 32; ++it) {
                    bool p = predT[(it * 32 + lane) * QT + q] != 0;
                    unsigned m = __builtin_amdgcn_ballot_w32(p);
                    if (m) {
                        int pre = __builtin_popcount(m & lt);
                        if (p && (cnt + pre) < NS_)
                            selIdx[q][cnt + pre] = cbase + it * 32 + lane;
                        cnt += __builtin_popcount(m);
                        if (cnt >= NS_) { cnt = NS_; break; }   // wave-uniform
                    }
                }
            }
            if (lane == 0 && was < NS_ && cnt == NS_) atomicAdd(&nfull, 1);
            if (sub) cnt1 = cnt; else cnt0 = cnt;
        }
        __syncthreads();   // predT reuse barrier; nfull writes visible
        if (nfull == QT) break;   // block-uniform early exit
    }

    if (lane == 0) { selCnt[wave * 2] = cnt0; selCnt[wave * 2 + 1] = cnt1; }
    __syncthreads();

    // ---- tail semantics: pad with first hit; if no hits at all, pad with 0
    if (tid < QT) {
        int c    = selCnt[tid];
        int fill = (c > 0) ? selIdx[tid][0] : 0;
        for (int k = c; k < NS_; ++k) selIdx[tid][k] = fill;
    }
    __syncthreads();

    // ===== phase C: gather + write output (B, 67, S, 32) =====
    const size_t chs = (size_t)S_ * NS_;         // channel stride
    for (int item = tid; item < QT * NS_; item += blockDim.x) {
        int sl = item >> 5;                      // local query
        int k  = item & 31;                      // sample slot (contiguous across lanes)
        int j  = selIdx[sl][k];

        const float* pp = xyz + ((size_t)b * N_ + j) * 3;
        float gx = pp[0] - qx[sl];
        float gy = pp[1] - qy[sl];
        float gz = pp[2] - qz[sl];

        size_t ob = (((size_t)b * 67) * (size_t)S_ + (size_t)(s0 + sl)) * NS_ + k;
        out[ob]           = gx;
        out[ob +     chs] = gy;
        out[ob + 2 * chs] = gz;

        const float4* fp = (const float4*)(feat + ((size_t)b * N_ + j) * (size_t)C_);
        size_t o = ob + 3 * chs;
        #pragma unroll
        for (int cc = 0; cc < 16; ++cc) {
            float4 f = fp[cc];
            out[o] = f.x; o += chs;
            out[o] = f.y; o += chs;
            out[o] = f.z; o += chs;
            out[o] = f.w; o += chs;
        }
    }
}

extern "C" void kernel_launch(void* const* d_in, const int* in_sizes, int n_in,
                              void* d_out, int out_size, void* d_ws, size_t ws_size,
                              hipStream_t stream) {
    (void)in_sizes; (void)n_in; (void)d_ws; (void)ws_size; (void)out_size;
    const float* xyz     = (const float*)d_in[0];
    const float* new_xyz = (const float*)d_in[1];
    const float* feat    = (const float*)d_in[2];
    float* out           = (float*)d_out;

    dim3 grid(B_ * (S_ / QT));   // 8 * 128 = 1024 blocks
    dim3 block(256);             // 8 waves
    qg_ballquery_wmma<<<grid, block, 0, stream>>>(xyz, new_xyz, feat, out);
}